// DynamicGCN_3453153706624
// MI455X (gfx1250) — compile-verified
//
#include <hip/hip_runtime.h>
#include <hip/hip_bf16.h>

// ---------------------------------------------------------------------------
// DynamicGCN for MI455X (gfx1250, wave32).
// Bandwidth-bound (~1.2 GB HBM traffic -> ~50us floor at 23.3 TB/s); dense
// GEMMs run on V_WMMA_F32_16X16X4_F32 to keep full fp32 numerics while still
// using the CDNA5 matrix pipe.
// ---------------------------------------------------------------------------

typedef __attribute__((ext_vector_type(2))) float v2f;
typedef __attribute__((ext_vector_type(8))) float v8f;

// ---------------------------------------------------------------------------
// C[M x 64] = A[M x K] @ B[K x 64] (+ optional bias), K % 4 == 0.
// One wave per 16-row strip, covering all 64 output columns (4 WMMA tiles).
// VGPR layouts per CDNA5 ISA 7.12.2:
//   A 16x4 f32 : lanes 0-15 -> M=lane, K={0,1} in v[0],v[1];
//                lanes 16-31 -> M=lane-16, K={2,3}.
//   B 4x16 f32 : lanes 0-15 -> N=lane, K={0,1}; lanes 16-31 -> K={2,3}.
//   C/D 16x16  : VGPR r, lanes 0-15 -> M=r, N=lane; lanes 16-31 -> M=8+r.
// ---------------------------------------------------------------------------
__global__ __launch_bounds__(256) void gemm_f32_wmma(
    const float* __restrict__ A, const float* __restrict__ B,
    const float* __restrict__ bias, float* __restrict__ C, int M, int K) {
  const int wave = threadIdx.x >> 5;
  const int lane = threadIdx.x & 31;
  const int half = lane >> 4;   // 0: K pair {0,1}; 1: K pair {2,3}
  const int l15  = lane & 15;

  const int row0 = (blockIdx.x * 8 + wave) * 16;
  if (row0 >= M) return;  // wave-uniform: EXEC stays all-ones for WMMA

  v8f acc0 = {}, acc1 = {}, acc2 = {}, acc3 = {};
  const float* aRow = A + (size_t)(row0 + l15) * K + 2 * half;

  for (int k = 0; k < K; k += 4) {
    v2f a;
    a.x = aRow[k];
    a.y = aRow[k + 1];
    const float* bRow = B + (size_t)(k + 2 * half) * 64 + l15;
    v2f b0, b1, b2, b3;
    b0.x = bRow[0];   b0.y = bRow[64];
    b1.x = bRow[16];  b1.y = bRow[80];
    b2.x = bRow[32];  b2.y = bRow[96];
    b3.x = bRow[48];  b3.y = bRow[112];
    acc0 = __builtin_amdgcn_wmma_f32_16x16x4_f32(false, a, false, b0,
                                                 (short)0, acc0, false, false);
    acc1 = __builtin_amdgcn_wmma_f32_16x16x4_f32(false, a, false, b1,
                                                 (short)0, acc1, false, false);
    acc2 = __builtin_amdgcn_wmma_f32_16x16x4_f32(false, a, false, b2,
                                                 (short)0, acc2, false, false);
    acc3 = __builtin_amdgcn_wmma_f32_16x16x4_f32(false, a, false, b3,
                                                 (short)0, acc3, false, false);
  }

  float bb0 = 0.f, bb1 = 0.f, bb2 = 0.f, bb3 = 0.f;
  if (bias) {
    bb0 = bias[l15];      bb1 = bias[l15 + 16];
    bb2 = bias[l15 + 32]; bb3 = bias[l15 + 48];
  }
#pragma unroll
  for (int r = 0; r < 8; ++r) {
    const int row = row0 + r + 8 * half;
    float* cRow = C + (size_t)row * 64 + l15;
    cRow[0]  = acc0[r] + bb0;
    cRow[16] = acc1[r] + bb1;
    cRow[32] = acc2[r] + bb2;
    cRow[48] = acc3[r] + bb3;
  }
}

// ------------------------- degree / normalization --------------------------
__global__ void fill_ones(float* __restrict__ p, int n) {
  int i = blockIdx.x * blockDim.x + threadIdx.x;
  if (i < n) p[i] = 1.0f;  // self-loop weight
}

__global__ void deg_accum(const int* __restrict__ dst,
                          const float* __restrict__ ew,
                          float* __restrict__ deg, int ne) {
  int e = blockIdx.x * blockDim.x + threadIdx.x;
  if (e < ne) atomicAdd(&deg[dst[e]], ew[e]);
}

__global__ void deg_to_dinv(float* __restrict__ p, int n) {
  int i = blockIdx.x * blockDim.x + threadIdx.x;
  if (i < n) {
    float d = p[i];
    p[i] = (d > 0.f) ? rsqrtf(d) : 0.f;
  }
}

// ------------------------- aggregation (A + I) -----------------------------
// Self-loop term: agg[i,:] = h[i,:] * dinv[i]^2  (also initializes agg).
__global__ void self_loop_init(const float* __restrict__ h,
                               const float* __restrict__ dinv,
                               float* __restrict__ agg, int total) {
  int t = blockIdx.x * blockDim.x + threadIdx.x;
  if (t >= total) return;
  float di = dinv[t >> 6];
  agg[t] = h[t] * di * di;
}

// 16 threads per edge, float4 per thread: 256B coalesced gather of h[src]
// and 256B coalesced atomic scatter into agg[dst].
__global__ __launch_bounds__(256) void edge_scatter(
    const float* __restrict__ h, const int* __restrict__ src,
    const int* __restrict__ dst, const float* __restrict__ ew,
    const float* __restrict__ dinv, float* __restrict__ agg, int ne) {
  long long t = (long long)blockIdx.x * blockDim.x + threadIdx.x;
  int e = (int)(t >> 4);
  if (e >= ne) return;
  int f0 = ((int)t & 15) * 4;
  int s = src[e], d = dst[e];
  float nrm = dinv[s] * ew[e] * dinv[d];
  const float4 hv = *(const float4*)(h + (size_t)s * 64 + f0);
  float* ad = agg + (size_t)d * 64 + f0;
  atomicAdd(ad + 0, hv.x * nrm);
  atomicAdd(ad + 1, hv.y * nrm);
  atomicAdd(ad + 2, hv.z * nrm);
  atomicAdd(ad + 3, hv.w * nrm);
}

__global__ void bias_act(float* __restrict__ v, const float* __restrict__ b,
                         int total, int relu) {
  int t = blockIdx.x * blockDim.x + threadIdx.x;
  if (t >= total) return;
  float x = v[t] + b[t & 63];
  v[t] = relu ? fmaxf(x, 0.f) : x;
}

// ---------------------------------------------------------------------------
extern "C" void kernel_launch(void* const* d_in, const int* in_sizes, int n_in,
                              void* d_out, int out_size, void* d_ws,
                              size_t ws_size, hipStream_t stream) {
  const float* x  = (const float*)d_in[0];   // [N,384]
  const int*   ei = (const int*)d_in[1];     // [2,E]
  const float* ew = (const float*)d_in[2];   // [E]
  const float* Wm = (const float*)d_in[3];   // [384,64]
  const float* bm = (const float*)d_in[4];   // [64]
  const float* W1 = (const float*)d_in[5];   // [64,64]
  const float* b1 = (const float*)d_in[6];   // [64]
  const float* W2 = (const float*)d_in[7];   // [64,64]
  const float* b2 = (const float*)d_in[8];   // [64]
  float* out = (float*)d_out;                // [N,64]

  const int NN = in_sizes[0] / 384;
  const int EE = in_sizes[2];
  const int* src = ei;
  const int* dst = ei + EE;

  // Workspace layout: dinv[N] | x64[N*64] | h[N*64] | agg[N*64]  (~77 MB)
  char* ws = (char*)d_ws;
  float* dinv = (float*)ws;
  size_t off  = ((size_t)NN * sizeof(float) + 255) & ~(size_t)255;
  float* x64  = (float*)(ws + off);
  float* hbuf = (float*)(ws + off + 1 * (size_t)NN * 64 * sizeof(float));
  float* agg  = (float*)(ws + off + 2 * (size_t)NN * 64 * sizeof(float));

  const int total = NN * 64;
  const dim3 blk(256);
  const int gTot  = (total + 255) / 256;
  const int gN    = (NN + 255) / 256;
  const int gE    = (EE + 255) / 256;
  const int gE16  = (int)(((long long)EE * 16 + 255) / 256);
  const int gGemm = ((NN + 15) / 16 + 7) / 8;  // 8 waves/block, 16 rows/wave

  // Symmetric normalization: dinv = rsqrt(1 + sum_{e: dst==i} w_e)
  fill_ones<<<gN, blk, 0, stream>>>(dinv, NN);
  deg_accum<<<gE, blk, 0, stream>>>(dst, ew, dinv, EE);
  deg_to_dinv<<<gN, blk, 0, stream>>>(dinv, NN);

  // Feature transform: x64 = x @ Wm + bm
  gemm_f32_wmma<<<gGemm, blk, 0, stream>>>(x, Wm, bm, x64, NN, 384);

  // Layer 1: agg = Dinv(A+I)Dinv (x64 @ W1); then + b1, ReLU
  gemm_f32_wmma<<<gGemm, blk, 0, stream>>>(x64, W1, nullptr, hbuf, NN, 64);
  self_loop_init<<<gTot, blk, 0, stream>>>(hbuf, dinv, agg, total);
  edge_scatter<<<gE16, blk, 0, stream>>>(hbuf, src, dst, ew, dinv, agg, EE);
  bias_act<<<gTot, blk, 0, stream>>>(agg, b1, total, 1);

  // Layer 2: out = Dinv(A+I)Dinv (agg @ W2) + b2
  gemm_f32_wmma<<<gGemm, blk, 0, stream>>>(agg, W2, nullptr, hbuf, NN, 64);
  self_loop_init<<<gTot, blk, 0, stream>>>(hbuf, dinv, out, total);
  edge_scatter<<<gE16, blk, 0, stream>>>(hbuf, src, dst, ew, dinv, out, EE);
  bias_act<<<gTot, blk, 0, stream>>>(out, b2, total, 0);
}